// CapsuleLayer_44126493999542
// MI455X (gfx1250) — compile-verified
//
#include <hip/hip_runtime.h>

// CDNA5 / gfx1250, wave32. Fully fused capsule routing:
//   priors recomputed on the fly via V_WMMA_F32_16X16X4_F32 (K=8 -> 2 chained WMMAs),
//   everything L2-resident, logits in LDS, zero workspace.

typedef __attribute__((ext_vector_type(2))) float v2f;
typedef __attribute__((ext_vector_type(8))) float v8f;

#define B_   256
#define N_   1152
#define CI_  8
#define C_   10
#define CO_  16
#define BT_  16                 // batch rows per workgroup (== WMMA M)
#define NW_  16                 // wave32s per workgroup
#define BLK_ (NW_ * 32)         // 512 threads
#define NSTRIPE_ (N_ / NW_)     // 72 route nodes per wave
#define LGP_ (N_ + 1)           // padded logits row stride (bank-conflict-free)

__global__ __launch_bounds__(BLK_) void caps_route_wmma(
    const float* __restrict__ x,     // [B, N, CI]
    const float* __restrict__ rw,    // [C, N, CI, CO]
    float* __restrict__ out)         // [C, B, CO]
{
  __shared__ float lg[BT_][LGP_];        // routing logits, per (b_local, n)
  __shared__ float sred[NW_][BT_ * CO_]; // per-wave partial weighted sums
  __shared__ float sv[BT_ * CO_];        // reduced s
  __shared__ float vv[BT_ * CO_];        // squashed v
  __shared__ float mxs[BT_];             // softmax row max
  __shared__ float isum[BT_];            // softmax 1/sum(exp)

  const int c    = blockIdx.x / (B_ / BT_);
  const int bt   = blockIdx.x % (B_ / BT_);
  const int b0   = bt * BT_;
  const int tid  = threadIdx.x;
  const int lane = tid & 31;
  const int wave = tid >> 5;
  const int half = lane >> 4;    // 0: lanes 0-15, 1: lanes 16-31
  const int m    = lane & 15;    // row (A M index) / col (B/C/D N index)

  // ---- init logits to zero ----
  for (int b = 0; b < BT_; ++b)
    for (int n = tid; n < N_; n += BLK_) lg[b][n] = 0.0f;
  __syncthreads();

  const int n0 = wave * NSTRIPE_;
  // A operand base: row m of x tile, K slice starts at i = 2*half
  const float* xrow = x + (size_t)(b0 + m) * N_ * CI_ + 2 * half;
  // B operand base: W[c, n, k=2*half + {0,1}(+4)], column o = m
  const float* wrow = rw + (size_t)c * N_ * CI_ * CO_ + (2 * half) * CO_ + m;

  for (int it = 0; it < 3; ++it) {
    // ---- softmax stats over n: wave w owns batch row b = w ----
    {
      const int bb = wave;
      float mx = -3.0e38f;
      for (int n = lane; n < N_; n += 32) mx = fmaxf(mx, lg[bb][n]);
      for (int off = 16; off >= 1; off >>= 1) mx = fmaxf(mx, __shfl_xor(mx, off, 32));
      float se = 0.0f;
      for (int n = lane; n < N_; n += 32) se += __expf(lg[bb][n] - mx);
      for (int off = 16; off >= 1; off >>= 1) se += __shfl_xor(se, off, 32);
      if (lane == 0) { mxs[bb] = mx; isum[bb] = 1.0f / se; }
    }
    __syncthreads();

    // ---- weighted-sum pass: WMMA accumulator = sum_n probs[b][n] * (X_n @ W_n) ----
    {
      const float rmx = mxs[m];
      const float ris = isum[m];
      v8f acc = {0.f, 0.f, 0.f, 0.f, 0.f, 0.f, 0.f, 0.f};
      for (int n = n0; n < n0 + NSTRIPE_; ++n) {
        const float prob = __expf(lg[m][n] - rmx) * ris;
        const float* xp = xrow + (size_t)n * CI_;
        const float* wp = wrow + (size_t)n * (CI_ * CO_);
        v2f A0 = { xp[0] * prob, xp[1] * prob };   // K = 2*half, 2*half+1
        v2f A1 = { xp[4] * prob, xp[5] * prob };   // K = 4+2*half, 5+2*half
        v2f B0 = { wp[0],        wp[CO_] };        // W[k][o=m], W[k+1][o=m]
        v2f B1 = { wp[4 * CO_],  wp[5 * CO_] };
        acc = __builtin_amdgcn_wmma_f32_16x16x4_f32(false, A0, false, B0,
                                                    (short)0, acc, false, false);
        acc = __builtin_amdgcn_wmma_f32_16x16x4_f32(false, A1, false, B1,
                                                    (short)0, acc, false, false);
      }
#pragma unroll
      for (int g = 0; g < 8; ++g)                 // C/D layout: row = g + 8*half, col = m
        sred[wave][(g + 8 * half) * CO_ + m] = acc[g];
    }
    __syncthreads();

    // ---- deterministic fixed-order reduce over waves, then squash ----
    if (tid < BT_ * CO_) {
      float s = 0.0f;
#pragma unroll
      for (int w = 0; w < NW_; ++w) s += sred[w][tid];
      sv[tid] = s;
    }
    __syncthreads();
    if (tid < BT_) {
      float sq = 0.0f;
#pragma unroll
      for (int o = 0; o < CO_; ++o) { float t = sv[tid * CO_ + o]; sq += t * t; }
      const float f = sqrtf(sq) / (1.0f + sq);    // squash factor: |s|/(1+|s|^2)
#pragma unroll
      for (int o = 0; o < CO_; ++o) vv[tid * CO_ + o] = sv[tid * CO_ + o] * f;
    }
    __syncthreads();

    // ---- agreement pass: logits[b][n] += <P_n[b,:], v[b,:]> (skip on last iter) ----
    if (it < 2) {
      for (int n = n0; n < n0 + NSTRIPE_; ++n) {
        const float* xp = xrow + (size_t)n * CI_;
        const float* wp = wrow + (size_t)n * (CI_ * CO_);
        v2f A0 = { xp[0], xp[1] };
        v2f A1 = { xp[4], xp[5] };
        v2f B0 = { wp[0],       wp[CO_] };
        v2f B1 = { wp[4 * CO_], wp[5 * CO_] };
        v8f p = {0.f, 0.f, 0.f, 0.f, 0.f, 0.f, 0.f, 0.f};
        p = __builtin_amdgcn_wmma_f32_16x16x4_f32(false, A0, false, B0,
                                                  (short)0, p, false, false);
        p = __builtin_amdgcn_wmma_f32_16x16x4_f32(false, A1, false, B1,
                                                  (short)0, p, false, false);
#pragma unroll
        for (int g = 0; g < 8; ++g) {
          const int row = g + 8 * half;
          float t = p[g] * vv[row * CO_ + m];
          // sum over the 16 output columns (stays within each 16-lane half)
          t += __shfl_xor(t, 1, 32);
          t += __shfl_xor(t, 2, 32);
          t += __shfl_xor(t, 4, 32);
          t += __shfl_xor(t, 8, 32);
          if (m == 0) lg[row][n] += t;            // lanes 0 and 16 write distinct rows
        }
      }
    }
    __syncthreads();
  }

  // ---- final outputs: [C, B, 1, 1, CO] flat = (c*B + b)*CO + o ----
  if (tid < BT_ * CO_) {
    const int bl = tid / CO_;
    const int o  = tid % CO_;
    out[((size_t)c * B_ + (b0 + bl)) * CO_ + o] = vv[tid];
  }
}

extern "C" void kernel_launch(void* const* d_in, const int* in_sizes, int n_in,
                              void* d_out, int out_size, void* d_ws, size_t ws_size,
                              hipStream_t stream) {
  const float* x  = (const float*)d_in[0];   // [256, 1152, 8] f32
  const float* rw = (const float*)d_in[1];   // [10, 1152, 8, 16] f32
  float* out = (float*)d_out;                // [10, 256, 16] f32

  dim3 grid(C_ * (B_ / BT_));                // 10 * 16 = 160 workgroups
  caps_route_wmma<<<grid, dim3(BLK_), 0, stream>>>(x, rw, out);

  (void)in_sizes; (void)n_in; (void)out_size; (void)d_ws; (void)ws_size;
}